// GCNProbe_18519898980624
// MI455X (gfx1250) — compile-verified
//
#include <hip/hip_runtime.h>

typedef __attribute__((ext_vector_type(2))) float v2f;
typedef __attribute__((ext_vector_type(8))) float v8f;

#define N_NODES  50000
#define N_EDGES  800000
#define D_IN     128
#define D_OUT    64
#define N_GRAPHS 512

// ---------------- degree / normalization ----------------

__global__ void k_deg_init(float* __restrict__ deg) {
    int i = blockIdx.x * blockDim.x + threadIdx.x;
    if (i < N_NODES) deg[i] = 1.0f;   // self-loop weight pre-seeded
}

__global__ void k_deg_edges(const int* __restrict__ ei,
                            const float* __restrict__ ew,
                            float* __restrict__ deg) {
    int e = blockIdx.x * blockDim.x + threadIdx.x;
    if (e < N_EDGES) {
        int dst = ei[N_EDGES + e];
        atomicAdd(&deg[dst], ew[e]);
    }
}

__global__ void k_dis(const float* __restrict__ deg, float* __restrict__ dis) {
    int i = blockIdx.x * blockDim.x + threadIdx.x;
    if (i < N_NODES) {
        float d = deg[i];
        dis[i] = (d > 0.0f) ? rsqrtf(d) : 0.0f;
    }
}

// ---------------- h = x @ W_conv^T  (WMMA f32 16x16x4) ----------------
// grid.x = 3125 M-tiles, block = 256 (8 waves); wave w owns N-tile w.

__global__ void k_gemm_h(const float* __restrict__ x,
                         const float* __restrict__ Wc,
                         float* __restrict__ h) {
    const int lane  = threadIdx.x & 31;
    const int wv    = threadIdx.x >> 5;     // N tile 0..7
    const int mtile = blockIdx.x;           // 0..3124
    const int half  = lane >> 4;
    const int l     = lane & 15;
    const int row   = mtile * 16 + l;       // A-matrix row (M)
    const int n     = wv * 16 + l;          // B-matrix col (N)

    v8f acc = {};
    #pragma unroll
    for (int k0 = 0; k0 < D_IN; k0 += 4) {
        const int ka = k0 + 2 * half;       // lanes 0-15: K=k0,k0+1; 16-31: k0+2,k0+3
        v2f a, b;
        a.x = x[row * D_IN + ka];
        a.y = x[row * D_IN + ka + 1];
        b.x = Wc[n * D_IN + ka];            // B[k][n] = W_conv[n][k]
        b.y = Wc[n * D_IN + ka + 1];
        acc = __builtin_amdgcn_wmma_f32_16x16x4_f32(
                  false, a, false, b, (short)0, acc, false, false);
    }
    #pragma unroll
    for (int r = 0; r < 8; ++r) {
        int m = mtile * 16 + half * 8 + r;  // C/D layout: VGPR r -> rows r / r+8
        h[m * D_IN + wv * 16 + l] = acc[r];
    }
}

// ---------------- edge gather-scale-scatter ----------------
// 32 lanes per edge; each lane moves 4 floats (float4 gather, 4 f32 atomics).

__global__ void k_aggregate(const int* __restrict__ ei,
                            const float* __restrict__ ew,
                            const float* __restrict__ dis,
                            const float* __restrict__ h,
                            float* __restrict__ out) {
    unsigned gid = blockIdx.x * blockDim.x + threadIdx.x;
    int e = (int)(gid >> 5);
    if (e >= N_EDGES) return;
    int lane = (int)(gid & 31);

    int src = ei[e];
    int dst = ei[N_EDGES + e];
    float nrm = dis[src] * ew[e] * dis[dst];

    const float4 v = *((const float4*)(h + (size_t)src * D_IN) + lane);
    float* o = out + (size_t)dst * D_IN + lane * 4;
    atomicAdd(o + 0, nrm * v.x);
    atomicAdd(o + 1, nrm * v.y);
    atomicAdd(o + 2, nrm * v.z);
    atomicAdd(o + 3, nrm * v.w);
}

// self-loop contribution + bias: runs after edge atomics (stream order),
// one thread per element -> plain read-modify-write, no races.
__global__ void k_selfloop_bias(const float* __restrict__ dis,
                                const float* __restrict__ h,
                                const float* __restrict__ bc,
                                float* __restrict__ out) {
    int gid = blockIdx.x * blockDim.x + threadIdx.x;
    if (gid >= N_NODES * D_IN) return;
    int i = gid >> 7;
    int c = gid & (D_IN - 1);
    float ds = dis[i];
    out[gid] += ds * ds * h[gid] + bc[c];
}

// ---------------- mean pool ----------------

__global__ void k_pool(const int* __restrict__ batch,
                       const float* __restrict__ out,
                       float* __restrict__ pooled) {
    int gid = blockIdx.x * blockDim.x + threadIdx.x;
    if (gid >= N_NODES * D_IN) return;
    int i = gid >> 7;
    int c = gid & (D_IN - 1);
    atomicAdd(&pooled[batch[i] * D_IN + c], out[gid]);
}

__global__ void k_count(const int* __restrict__ batch, float* __restrict__ cnt) {
    int i = blockIdx.x * blockDim.x + threadIdx.x;
    if (i < N_NODES) atomicAdd(&cnt[batch[i]], 1.0f);
}

__global__ void k_invc(const float* __restrict__ cnt, float* __restrict__ invc) {
    int g = blockIdx.x * blockDim.x + threadIdx.x;
    if (g < N_GRAPHS) invc[g] = 1.0f / fmaxf(cnt[g], 1.0f);
}

// ---------------- y = (pooled/cnt) @ W_lin^T + b_lin  (WMMA) ----------------
// grid.x = 32 M-tiles, block = 128 (4 waves); wave w owns N-tile w (64 = 4x16).

__global__ void k_gemm_out(const float* __restrict__ pooled,
                           const float* __restrict__ invc,
                           const float* __restrict__ Wl,
                           const float* __restrict__ bl,
                           float* __restrict__ y) {
    const int lane  = threadIdx.x & 31;
    const int wv    = threadIdx.x >> 5;     // N tile 0..3
    const int mtile = blockIdx.x;           // 0..31
    const int half  = lane >> 4;
    const int l     = lane & 15;
    const int g     = mtile * 16 + l;
    const int n     = wv * 16 + l;
    const float ic  = invc[g];

    v8f acc = {};
    #pragma unroll
    for (int k0 = 0; k0 < D_IN; k0 += 4) {
        const int ka = k0 + 2 * half;
        v2f a, b;
        a.x = pooled[g * D_IN + ka] * ic;   // fused mean
        a.y = pooled[g * D_IN + ka + 1] * ic;
        b.x = Wl[n * D_IN + ka];            // B[k][o] = W_lin[o][k]
        b.y = Wl[n * D_IN + ka + 1];
        acc = __builtin_amdgcn_wmma_f32_16x16x4_f32(
                  false, a, false, b, (short)0, acc, false, false);
    }
    #pragma unroll
    for (int r = 0; r < 8; ++r) {
        int gg = mtile * 16 + half * 8 + r;
        int oo = wv * 16 + l;
        y[gg * D_OUT + oo] = acc[r] + bl[oo];
    }
}

// ---------------- launcher ----------------

extern "C" void kernel_launch(void* const* d_in, const int* in_sizes, int n_in,
                              void* d_out, int out_size, void* d_ws, size_t ws_size,
                              hipStream_t stream) {
    const float* x     = (const float*)d_in[0];
    const int*   ei    = (const int*)  d_in[1];   // [2, N_EDGES]
    const int*   batch = (const int*)  d_in[2];
    const float* ew    = (const float*)d_in[3];
    const float* Wc    = (const float*)d_in[4];
    const float* bc    = (const float*)d_in[5];
    const float* Wl    = (const float*)d_in[6];
    const float* bl    = (const float*)d_in[7];
    float* y = (float*)d_out;

    char* ws = (char*)d_ws;
    size_t off = 0;
    auto take = [&](size_t bytes) -> float* {
        float* p = (float*)(ws + off);
        off += (bytes + 255) & ~((size_t)255);
        return p;
    };
    float* deg    = take((size_t)N_NODES * 4);
    float* dis    = take((size_t)N_NODES * 4);
    float* h      = take((size_t)N_NODES * D_IN * 4);
    float* agg    = take((size_t)N_NODES * D_IN * 4);
    float* pooled = take((size_t)N_GRAPHS * D_IN * 4);
    float* cnt    = take((size_t)N_GRAPHS * 4);
    float* invc   = take((size_t)N_GRAPHS * 4);
    (void)ws_size; (void)in_sizes; (void)n_in; (void)out_size;

    hipMemsetAsync(agg,    0, (size_t)N_NODES * D_IN * 4, stream);
    hipMemsetAsync(pooled, 0, (size_t)N_GRAPHS * D_IN * 4, stream);
    hipMemsetAsync(cnt,    0, (size_t)N_GRAPHS * 4, stream);

    k_deg_init <<<(N_NODES + 255) / 256, 256, 0, stream>>>(deg);
    k_deg_edges<<<(N_EDGES + 255) / 256, 256, 0, stream>>>(ei, ew, deg);
    k_dis      <<<(N_NODES + 255) / 256, 256, 0, stream>>>(deg, dis);

    k_gemm_h<<<N_NODES / 16, 256, 0, stream>>>(x, Wc, h);       // 3125 blocks

    k_aggregate<<<(N_EDGES * 32) / 256, 256, 0, stream>>>(ei, ew, dis, h, agg);
    k_selfloop_bias<<<(N_NODES * D_IN + 255) / 256, 256, 0, stream>>>(dis, h, bc, agg);

    k_pool <<<(N_NODES * D_IN + 255) / 256, 256, 0, stream>>>(batch, agg, pooled);
    k_count<<<(N_NODES + 255) / 256, 256, 0, stream>>>(batch, cnt);
    k_invc <<<(N_GRAPHS + 255) / 256, 256, 0, stream>>>(cnt, invc);

    k_gemm_out<<<N_GRAPHS / 16, 128, 0, stream>>>(pooled, invc, Wl, bl, y);
}